// KVCache_9603546874180
// MI455X (gfx1250) — compile-verified
//
#include <hip/hip_runtime.h>

// ---------------------------------------------------------------------------
// KV-cache update for MI455X (gfx1250).
//   out_k = copy(k); out_v = copy(v); out_{k,v}[:, :, index[j], :] = {k,v}_val
// Pure bandwidth problem: ~2 GiB streamed -> ~92 us floor at 23.3 TB/s.
//
// Data-movement path chosen for the chip:
//  * global_load_b128 / global_store_b128 with TH=NT cache hints
//    (2 GiB stream >> 192 MB L2; don't thrash it)
//  * GVS addressing: SGPR64 base + 32-bit VGPR offset (all byte offsets
//    < 2^29), so the inner loop does 32-bit VALU address math only
//  * global_prefetch_b8 one batch ahead to prime GL2 for the NT loads
// Scatter of the 16 new rows runs afterwards via stream ordering.
// ---------------------------------------------------------------------------

typedef float v4f __attribute__((ext_vector_type(4)));

constexpr unsigned Bc = 8, Hc = 32, Sc = 4096, Dc = 128, SN = 16;
constexpr size_t   PLANE = (size_t)Bc * Hc * Sc * Dc;     // 134,217,728 floats
constexpr size_t   N4    = PLANE / 4;                     //  33,554,432 float4

constexpr unsigned BLOCKS   = 8192, TPB = 256;
constexpr unsigned THREADS  = BLOCKS * TPB;               // 2,097,152
constexpr unsigned PER_THR  = (unsigned)(N4 / THREADS);   // 16 float4 per plane
constexpr unsigned BATCH    = 4;
constexpr unsigned ITERS    = PER_THR / BATCH;            // 4
constexpr unsigned STEP     = THREADS * 16u;              // bytes between lanes' consecutive elems (32 MiB)

static_assert(N4 % THREADS == 0, "exact tiling");

// ---- bulk streaming copy: both planes in one kernel, NT b128, u32 offsets --
__global__ __launch_bounds__(TPB) void kv_copy_nt(const float* __restrict__ k,
                                                  const float* __restrict__ v,
                                                  float* __restrict__ ok,
                                                  float* __restrict__ ov) {
  const char* ks = (const char*)k;
  const char* vs = (const char*)v;
  char*       kd = (char*)ok;
  char*       vd = (char*)ov;

  unsigned tid = blockIdx.x * TPB + threadIdx.x;
  unsigned off = tid * 16u;                 // < 32 MiB, fits u32 easily

#pragma unroll
  for (unsigned it = 0; it < ITERS; ++it) {
    unsigned base = off + it * (BATCH * STEP);   // max < 2^29, u32-safe

    if (it + 1 < ITERS) {                        // prime GL2 one batch ahead
      __builtin_prefetch(ks + base + BATCH * STEP, 0, 0);
      __builtin_prefetch(vs + base + BATCH * STEP, 0, 0);
    }

    // 8 independent B128 NT loads in flight (4 from k, 4 from v) ...
    v4f a0 = __builtin_nontemporal_load((const v4f*)(ks + base + 0 * STEP));
    v4f a1 = __builtin_nontemporal_load((const v4f*)(ks + base + 1 * STEP));
    v4f a2 = __builtin_nontemporal_load((const v4f*)(ks + base + 2 * STEP));
    v4f a3 = __builtin_nontemporal_load((const v4f*)(ks + base + 3 * STEP));
    v4f b0 = __builtin_nontemporal_load((const v4f*)(vs + base + 0 * STEP));
    v4f b1 = __builtin_nontemporal_load((const v4f*)(vs + base + 1 * STEP));
    v4f b2 = __builtin_nontemporal_load((const v4f*)(vs + base + 2 * STEP));
    v4f b3 = __builtin_nontemporal_load((const v4f*)(vs + base + 3 * STEP));
    // ... then 8 NT stores
    __builtin_nontemporal_store(a0, (v4f*)(kd + base + 0 * STEP));
    __builtin_nontemporal_store(a1, (v4f*)(kd + base + 1 * STEP));
    __builtin_nontemporal_store(a2, (v4f*)(kd + base + 2 * STEP));
    __builtin_nontemporal_store(a3, (v4f*)(kd + base + 3 * STEP));
    __builtin_nontemporal_store(b0, (v4f*)(vd + base + 0 * STEP));
    __builtin_nontemporal_store(b1, (v4f*)(vd + base + 1 * STEP));
    __builtin_nontemporal_store(b2, (v4f*)(vd + base + 2 * STEP));
    __builtin_nontemporal_store(b3, (v4f*)(vd + base + 3 * STEP));
  }
}

// ---- scatter of the S_NEW updated rows -------------------------------------
// One wave32 per (b, h, j) row: 32 lanes x float4 = 128 floats = one D-row.
__global__ __launch_bounds__(TPB) void kv_scatter(const float* __restrict__ kval,
                                                  const float* __restrict__ vval,
                                                  const int* __restrict__ index,
                                                  float* __restrict__ outk,
                                                  float* __restrict__ outv) {
  unsigned tid  = blockIdx.x * TPB + threadIdx.x;
  unsigned wave = tid >> 5;            // one row per wave32
  unsigned lane = threadIdx.x & 31u;
  if (wave >= Bc * Hc * SN) return;

  unsigned j  = wave & (SN - 1);       // SN == 16
  unsigned bh = wave >> 4;
  unsigned s  = (unsigned)index[j];    // scatter destination along S

  // byte offsets; max dst offset ~2^29, u32-safe -> GVS addressing
  unsigned srcOff = ((bh * SN + j) * (Dc / 4) + lane) * 16u;
  unsigned dstOff = ((bh * Sc + s) * (Dc / 4) + lane) * 16u;

  v4f a = *(const v4f*)((const char*)kval + srcOff);
  v4f b = *(const v4f*)((const char*)vval + srcOff);
  *(v4f*)((char*)outk + dstOff) = a;
  *(v4f*)((char*)outv + dstOff) = b;
}

extern "C" void kernel_launch(void* const* d_in, const int* in_sizes, int n_in,
                              void* d_out, int out_size, void* d_ws, size_t ws_size,
                              hipStream_t stream) {
  (void)in_sizes; (void)n_in; (void)out_size; (void)d_ws; (void)ws_size;

  const float* k     = (const float*)d_in[0];
  const float* v     = (const float*)d_in[1];
  const float* k_val = (const float*)d_in[2];
  const float* v_val = (const float*)d_in[3];
  const int*   index = (const int*)d_in[4];

  float* out_k = (float*)d_out;
  float* out_v = (float*)d_out + PLANE;

  // Single copy launch covers both planes: 2M threads x (16+16) float4.
  kv_copy_nt<<<dim3(BLOCKS), dim3(TPB), 0, stream>>>(k, v, out_k, out_v);

  // B*H*S_NEW = 4096 rows -> 4096 waves -> 131,072 threads -> 512 blocks.
  // Runs after the copy by stream ordering, so indexed rows end up with the
  // new values (matching .at[...].set semantics).
  unsigned rows    = Bc * Hc * SN;
  unsigned threads = rows * 32u;
  kv_scatter<<<dim3(threads / TPB), dim3(TPB), 0, stream>>>(
      k_val, v_val, index, out_k, out_v);
}